// Model_60713657696948
// MI455X (gfx1250) — compile-verified
//
#include <hip/hip_runtime.h>

typedef __attribute__((ext_vector_type(2))) float v2f;
typedef __attribute__((ext_vector_type(8))) float v8f;

#define BM 128
#define BN 128
#define BK 16
#define LDA (BK + 4)        // 20 dwords: conflict-free b64 A-frag loads
#define LDB2 (2 * BN + 32)  // 288 dwords: K-pair-interleaved B rows, +32 puts
                            // the h=1 half-wave on the other 32 banks
#define NTHREADS 256

__global__ __launch_bounds__(NTHREADS, 1)
void grouped_gemm_f32_wmma(const float* __restrict__ A,
                           const float* __restrict__ B,
                           const int* __restrict__ group_list,
                           float* __restrict__ Out,
                           int M, int K, int N, int G)
{
    __shared__ float sA[2][BM * LDA];          // [m][k]        row-major
    __shared__ float sB[2][(BK / 2) * LDB2];   // [k/2][n][2]   K-pair interleaved

    const int tid  = threadIdx.x;
    const int row0 = blockIdx.y * BM;
    const int col0 = blockIdx.x * BN;

    // group of this row tile (equal-size groups: 128-row tiles never straddle)
    int g = 0;
    while (g < G - 1 && group_list[g] <= row0) ++g;
    const float* __restrict__ Bg = B + (size_t)g * K * N;

    const int NKB = K / BK;    // 32, even

    // ---- global->reg staging: incrementing pointers + immediate offsets ----
    // A tile: 128 rows x 16 cols; 4 threads/row (float4), rows amRow and amRow+64
    const int amRow = tid >> 2;          // 0..63
    const int amCol = (tid & 3) * 4;     // 0,4,8,12
    // B tile: 16 rows x 128 cols as 2x2 micro-blocks, 2 per thread.
    //   p=0: (k2, n2);  p=1: (k2+4, n2)  ->  global disp is exactly +8*N
    const int k2 = tid >> 6;             // 0..3
    const int n2 = (tid & 63) << 1;      // 0..126

    // loop-invariant LDS staging indices (immediates after addressing)
    const int sAidx = amRow * LDA + amCol;
    const int sBidx = k2 * LDB2 + n2 * 2;

    const float* aPtr = A + (size_t)(row0 + amRow) * K + amCol;
    const float* bPtr = Bg + (size_t)(2 * k2) * N + col0 + n2;

    float4 aReg[2], bReg[2];

    auto loadGlobal = [&](const float* ap, const float* bp) {
        aReg[0] = *(const float4*)(ap);
        aReg[1] = *(const float4*)(ap + (size_t)64 * K);   // imm offset 128 KB
        v2f r00 = *(const v2f*)(bp);                       // row 2k2   : n2,n2+1
        v2f r01 = *(const v2f*)(bp + N);                   // row 2k2+1
        v2f r10 = *(const v2f*)(bp + 8 * N);               // row 2k2+8
        v2f r11 = *(const v2f*)(bp + 9 * N);               // row 2k2+9
        // interleave so (K, K+1) are adjacent per column
        bReg[0] = make_float4(r00.x, r01.x, r00.y, r01.y);
        bReg[1] = make_float4(r10.x, r11.x, r10.y, r11.y);
    };
    auto storeLds = [&](int buf) {
        *(float4*)&sA[buf][sAidx]            = aReg[0];
        *(float4*)&sA[buf][sAidx + 64 * LDA] = aReg[1];
        *(float4*)&sB[buf][sBidx]            = bReg[0];
        *(float4*)&sB[buf][sBidx + 4 * LDB2] = bReg[1];
    };

    // ---- wave tiling: 8 waves as 4(M) x 2(N); wave patch = 32x64 ----
    const int lane = tid & 31;
    const int w    = tid >> 5;
    const int wm   = w & 3;              // 0..3
    const int wn   = w >> 2;             // 0..1
    const int h    = lane >> 4;          // 0/1
    const int lm   = lane & 15;          // 0..15

    v8f acc[2][4];
#pragma unroll
    for (int i = 0; i < 2; ++i)
#pragma unroll
        for (int j = 0; j < 4; ++j)
            acc[i][j] = (v8f){0.f, 0.f, 0.f, 0.f, 0.f, 0.f, 0.f, 0.f};

    auto compute = [&](int buf) {
        const float* As = sA[buf];
        const float* Bs = sB[buf];
#pragma unroll
        for (int kk = 0; kk < BK; kk += 4) {
            // A frags: lane lm = row M, VGPR r holds K = kk + r + 2h  -> b64 pair
            v2f af[2];
#pragma unroll
            for (int i = 0; i < 2; ++i) {
                const int r = (wm * 32 + i * 16 + lm) * LDA + kk + 2 * h;
                af[i] = *(const v2f*)&As[r];
            }
            // B frags: [k2][n][2] layout -> single b64 yields (K, K+1) pair
            const float* brow = &Bs[((kk >> 1) + h) * LDB2];
            v2f bf[4];
#pragma unroll
            for (int j = 0; j < 4; ++j) {
                const int c = wn * 64 + j * 16 + lm;
                bf[j] = *(const v2f*)&brow[c * 2];
            }
#pragma unroll
            for (int i = 0; i < 2; ++i)
#pragma unroll
                for (int j = 0; j < 4; ++j)
                    acc[i][j] = __builtin_amdgcn_wmma_f32_16x16x4_f32(
                        false, af[i], false, bf[j],
                        (short)0, acc[i][j], false, false);
        }
    };

    // ---- pipelined main loop, unrolled x2: compile-time buffer indices ----
    loadGlobal(aPtr, bPtr);                 // stage kb=0
    storeLds(0);
    __syncthreads();
    aPtr += BK;                             // +64 B per K-chunk
    bPtr += (size_t)BK * N;                 // +32 KB per K-chunk

#pragma unroll 1
    for (int kb = 0; kb < NKB; kb += 2) {
        // stage kb+1 into buf1 while computing buf0 (= kb)
        loadGlobal(aPtr, bPtr);
        aPtr += BK;
        bPtr += (size_t)BK * N;
        compute(0);
        storeLds(1);
        __syncthreads();

        // stage kb+2 into buf0 while computing buf1 (= kb+1)
        const bool hasNext = (kb + 2 < NKB);
        if (hasNext) {
            loadGlobal(aPtr, bPtr);
            aPtr += BK;
            bPtr += (size_t)BK * N;
        }
        compute(1);
        if (hasNext) {
            storeLds(0);
            __syncthreads();
        }
    }

    // ---- epilogue: C/D layout -> global (row = base + jj + 8h, col = lm) ----
#pragma unroll
    for (int i = 0; i < 2; ++i) {
        const int rbase = row0 + wm * 32 + i * 16 + 8 * h;
#pragma unroll
        for (int j = 0; j < 4; ++j) {
            const int c = col0 + wn * 64 + j * 16 + lm;
#pragma unroll
            for (int jj = 0; jj < 8; ++jj)
                Out[(size_t)(rbase + jj) * N + c] = acc[i][j][jj];
        }
    }
}

extern "C" void kernel_launch(void* const* d_in, const int* in_sizes, int n_in,
                              void* d_out, int out_size, void* d_ws, size_t ws_size,
                              hipStream_t stream) {
    const float* a  = (const float*)d_in[0];
    const float* b  = (const float*)d_in[1];
    const int*   gl = (const int*)d_in[2];
    float* out = (float*)d_out;

    const int K = 512;
    const int N = 512;
    const int G = in_sizes[2];
    const int M = in_sizes[0] / K;

    dim3 grid(N / BN, M / BM);   // (4, 128)
    grouped_gemm_f32_wmma<<<grid, NTHREADS, 0, stream>>>(a, b, gl, out, M, K, N, G);
}